// intraclass_loss_41257455845383
// MI455X (gfx1250) — compile-verified
//
#include <hip/hip_runtime.h>
#include <hip/hip_bf16.h>

typedef __attribute__((ext_vector_type(2))) float v2f;
typedef __attribute__((ext_vector_type(8))) float v8f;

#define PAIRS_PER_TILE 15
#define TPB 256 /* 8 waves of 32 */

// ---------------------------------------------------------------------------
// Kernel 1: stable partition of binary target -> order[] (sorted pos -> row),
// plus n0 = count of zeros. Single 1024-thread block, LDS Hillis-Steele scan.
// ---------------------------------------------------------------------------
__global__ void icl_build_order(const int* __restrict__ target, int B,
                                int* __restrict__ order, int* __restrict__ n0_out) {
  __shared__ int scan[1024];
  const int tid = threadIdx.x;
  const int E = (B + 1023) >> 10;  // elements per thread
  const int g0 = tid * E;
  int cnt = 0;
  for (int e = 0; e < E; ++e) {
    int g = g0 + e;
    if (g < B && target[g] == 0) cnt++;
  }
  scan[tid] = cnt;
  __syncthreads();
  for (int off = 1; off < 1024; off <<= 1) {
    int v = (tid >= off) ? scan[tid - off] : 0;
    __syncthreads();
    scan[tid] += v;
    __syncthreads();
  }
  const int n0 = scan[1023];
  int zb = scan[tid] - cnt;  // zeros strictly before element g0
  for (int e = 0; e < E; ++e) {
    int g = g0 + e;
    if (g < B) {
      if (target[g] == 0) { order[zb] = g; zb++; }
      else                { order[n0 + g - zb] = g; }
    }
  }
  if (tid == 0) n0_out[0] = n0;
}

// ---------------------------------------------------------------------------
// Kernel 2: per tile of 16 consecutive sorted rows (stride 15 -> overlap 1),
// compute the 16x16 Gram matrix G = X_tile * X_tile^T with
// v_wmma_f32_16x16x4_f32, plus raw row sums. Emit:
//   pairdot[n]  = G[m][m+1]  (dot of sorted rows n, n+1)
//   rowsumsq[n] = G[m][m]
//   rowsum[n]   = sum of row n
// Fragment trick: for A (16x4, MxK) lane L holds row (L&15),
// cols k + 2*(L>>4) + {0,1}; the B fragment (4x16, KxN) of X^T has the
// identical lane layout, so one v2f load serves both operands.
// ---------------------------------------------------------------------------
__global__ void icl_gram_tile(const float* __restrict__ X,
                              const int* __restrict__ order, int B, int Dc,
                              float* __restrict__ rowsum,
                              float* __restrict__ rowsumsq,
                              float* __restrict__ pairdot) {
  __shared__ float g_lds[256];
  __shared__ float rs_lds[16];
  __shared__ int rows_lds[16];
  const int tid  = threadIdx.x;
  const int wave = tid >> 5;
  const int lane = tid & 31;
  const int t    = blockIdx.x;
  const int r0   = t * PAIRS_PER_TILE;

  g_lds[tid] = 0.0f;
  if (tid < 16) {
    rs_lds[tid] = 0.0f;
    int r = r0 + tid;
    if (r > B - 1) r = B - 1;  // clamp (harmless duplicate read)
    rows_lds[tid] = order[r];
  }
  __syncthreads();

  const int m     = lane & 15;
  const int khalf = (lane >> 4) << 1;  // 0 or 2
  const int kchunk = Dc >> 3;          // 8 waves split the columns
  const float* p = X + (size_t)rows_lds[m] * (size_t)Dc
                     + (size_t)(wave * kchunk + khalf);

  v8f acc = {};
  float rs = 0.0f;
  for (int k = 0; k < kchunk; k += 4) {
    v2f a = *(const v2f*)p;  // global_load_b64, 8B-aligned
    p += 4;
    rs += a.x + a.y;
    // D = A * B + C with A-frag == B-frag  => 16x16 Gram accumulation
    acc = __builtin_amdgcn_wmma_f32_16x16x4_f32(
        /*neg_a=*/false, a, /*neg_b=*/false, a,
        /*c_mod=*/(short)0, acc, /*reuse_a=*/false, /*reuse_b=*/false);
  }

  // D layout: VGPR g, lanes 0-15 -> (M=g, N=lane); lanes 16-31 -> (M=g+8).
#pragma unroll
  for (int g = 0; g < 8; ++g) {
    int M = g + ((lane >> 4) << 3);
    atomicAdd(&g_lds[M * 16 + (lane & 15)], acc[g]);
  }
  atomicAdd(&rs_lds[m], rs);
  __syncthreads();

  if (tid < 16) {
    int r = r0 + tid;
    if (tid < 15) {
      if (r + 1 <= B - 1) pairdot[r] = g_lds[tid * 16 + tid + 1];
      if (r <= B - 1) {
        rowsum[r]   = rs_lds[tid];
        rowsumsq[r] = g_lds[tid * 17];
      }
    } else if (r == B - 1) {  // tail row owned by the last tile
      rowsum[r]   = rs_lds[15];
      rowsumsq[r] = g_lds[255];
    }
  }
}

// ---------------------------------------------------------------------------
// Kernel 3: combine into the scalar (p0 - p1)^2.
//   css_i = Q_i - S_i^2/D ; std_i = sqrt(css_i/(D-1))
//   cov_n = dot_n - S_n*S_{n+1}/D ; corr_n = cov_n*(D-1)/sqrt(css_n*css_{n+1})
//   pair n same-class-0 iff n+1 < n0 ; same-class-1 iff n >= n0
// ---------------------------------------------------------------------------
__global__ void icl_finalize(const float* __restrict__ rowsum,
                             const float* __restrict__ rowsumsq,
                             const float* __restrict__ pairdot,
                             const int* __restrict__ n0p, int B, int Dc,
                             float* __restrict__ out) {
  __shared__ float s0_lds[256];
  __shared__ float s1_lds[256];
  const int tid = threadIdx.x;
  const int n0 = n0p[0];
  const float Dn = (float)Dc;
  float s0 = 0.0f, s1 = 0.0f;
  for (int n = tid; n < B - 1; n += 256) {
    bool c0 = (n + 1) < n0;
    bool c1 = n >= n0;
    if (c0 || c1) {
      float Sa = rowsum[n], Sb = rowsum[n + 1];
      float Qa = rowsumsq[n], Qb = rowsumsq[n + 1];
      float cssa = Qa - Sa * Sa / Dn;
      float cssb = Qb - Sb * Sb / Dn;
      float cov  = pairdot[n] - Sa * Sb / Dn;
      float corr = cov * (Dn - 1.0f) / sqrtf(cssa * cssb);
      if (c0) s0 += corr; else s1 += corr;
    }
  }
  s0_lds[tid] = s0;
  s1_lds[tid] = s1;
  __syncthreads();
  for (int off = 128; off > 0; off >>= 1) {
    if (tid < off) {
      s0_lds[tid] += s0_lds[tid + off];
      s1_lds[tid] += s1_lds[tid + off];
    }
    __syncthreads();
  }
  if (tid == 0) {
    int n1 = B - n0;
    float p0 = (n0 > 0) ? s0_lds[0] / (float)n0 : 0.0f;
    float p1 = (n1 > 0) ? s1_lds[0] / (float)n1 : 0.0f;
    float d = p0 - p1;
    out[0] = d * d;
  }
}

extern "C" void kernel_launch(void* const* d_in, const int* in_sizes, int n_in,
                              void* d_out, int out_size, void* d_ws, size_t ws_size,
                              hipStream_t stream) {
  const float* X      = (const float*)d_in[0];
  const int*   target = (const int*)d_in[1];
  const int B  = in_sizes[1];
  const int Dc = in_sizes[0] / B;

  int*   order    = (int*)d_ws;          // B ints
  int*   n0p      = order + B;           // 1 int (+pad)
  float* rowsum   = (float*)(n0p + 16);  // B floats (sorted-position indexed)
  float* rowsumsq = rowsum + B;          // B floats
  float* pairdot  = rowsumsq + B;        // B-1 floats

  icl_build_order<<<1, 1024, 0, stream>>>(target, B, order, n0p);

  const int ntiles = (B - 1 + PAIRS_PER_TILE - 1) / PAIRS_PER_TILE;
  icl_gram_tile<<<ntiles, TPB, 0, stream>>>(X, order, B, Dc,
                                            rowsum, rowsumsq, pairdot);

  icl_finalize<<<1, 256, 0, stream>>>(rowsum, rowsumsq, pairdot, n0p, B, Dc,
                                      (float*)d_out);
}